// VectorQuantizer_44358422233166
// MI455X (gfx1250) — compile-verified
//
#include <hip/hip_runtime.h>

typedef float v2f __attribute__((ext_vector_type(2)));
typedef float v8f __attribute__((ext_vector_type(8)));

#define N_CODES   512
#define CODE_DIM  64
#define HW        1024          // 32*32
#define NBATCH    256
#define N_TOT     (NBATCH * HW)             // 262144 z-vectors
#define NZ        (NBATCH * CODE_DIM * HW)  // 16777216 floats of z_q
#define ROWS_PB   64
#define THREADS   128
#define NBLOCKS   (N_TOT / ROWS_PB)         // 4096

// ---------------------------------------------------------------------------
// Setup: cnorm[i] = 0.5 * ||codebook[i]||^2   (512 floats into d_ws[0..511])
// ---------------------------------------------------------------------------
__global__ void vq_setup_44358422233166(const float* __restrict__ codebook,
                                        float* __restrict__ cnorm) {
  int i = threadIdx.x;                  // launched with 512 threads
  const float* row = codebook + (size_t)i * CODE_DIM;
  float s = 0.f;
#pragma unroll
  for (int c = 0; c < CODE_DIM; c += 4) {
    float4 v = *(const float4*)(row + c);
    s += v.x * v.x + v.y * v.y + v.z * v.z + v.w * v.w;
  }
  cnorm[i] = 0.5f * s;
}

// ---------------------------------------------------------------------------
// Main: per block = 64 z-rows. WMMA f32 16x16x4, A = codebook tile (16 codes x
// 4 ch), B = z tile (4 ch x 16 z). Lane-local argmax of (dot - 0.5||c||^2).
// ---------------------------------------------------------------------------
__global__ __launch_bounds__(THREADS)
void vq_main_44358422233166(const float* __restrict__ z,
                            const float* __restrict__ codebook,
                            const float* __restrict__ cnorm_g,
                            float* __restrict__ out,
                            float* __restrict__ partial) {
  __shared__ float z_lds[CODE_DIM * 65];      // [c][n], pad 65 vs bank conflicts
  __shared__ float cn_lds[N_CODES];
  __shared__ int   idx_lds[ROWS_PB];
  __shared__ float wsum[THREADS / 32];

  const int t    = threadIdx.x;
  const int n0   = blockIdx.x * ROWS_PB;      // first z-row of this block
  const int b    = n0 >> 10;                  // image index (n0 / 1024)
  const int rem0 = n0 & 1023;                 // h*32 + w base (64-aligned)
  const size_t zbase = (size_t)b * (CODE_DIM * HW) + rem0;

  // Front-load the 128KB codebook into cache (global_prefetch_b8); it is
  // reused by every block and must stay resident while z streams past it.
  {
    const char* cb8 = (const char*)codebook;
    for (int i = t; i < (N_CODES * CODE_DIM * 4) / 128; i += THREADS)
      __builtin_prefetch(cb8 + (size_t)i * 128, 0, 3);
  }

  // Stage z tile (streamed once -> non-temporal): contiguous over n for fixed
  // c -> coalesced.
  for (int i = t; i < CODE_DIM * ROWS_PB; i += THREADS) {
    int c = i >> 6, nl = i & 63;
    z_lds[c * 65 + nl] =
        __builtin_nontemporal_load(&z[zbase + (size_t)c * HW + nl]);
  }
  for (int i = t; i < N_CODES; i += THREADS) cn_lds[i] = cnorm_g[i];
  __syncthreads();

  const int wave = t >> 5;
  const int lane = t & 31;
  const int hi   = lane >> 4;                 // lane half: 0 or 1
  const int l15  = lane & 15;
  const int nl   = wave * 16 + l15;           // z column this lane owns in D
  const int koff = hi * 2;                    // K offset for A/B fragments

  float best = -3.4e38f;
  int   bidx = 0;

#pragma unroll 2
  for (int ct = 0; ct < N_CODES / 16; ++ct) {
    v8f acc = {0.f, 0.f, 0.f, 0.f, 0.f, 0.f, 0.f, 0.f};
    const float* crow = codebook + (size_t)(ct * 16 + l15) * CODE_DIM + koff;
#pragma unroll
    for (int s = 0; s < CODE_DIM / 4; ++s) {
      // A fragment: codebook[ct*16 + l15][4s+koff, 4s+koff+1]  (8B load)
      v2f a = *(const v2f*)(crow + 4 * s);
      // B fragment: z_lds[4s+koff][nl], z_lds[4s+koff+1][nl]  (loop-invariant
      // over ct -> compiler hoists all 32 values into registers)
      v2f bf;
      bf.x = z_lds[(4 * s + koff) * 65 + nl];
      bf.y = z_lds[(4 * s + koff + 1) * 65 + nl];
      acc = __builtin_amdgcn_wmma_f32_16x16x4_f32(
          /*neg_a=*/false, a, /*neg_b=*/false, bf,
          /*c_mod=*/(short)0, acc, /*reuse_a=*/false, /*reuse_b=*/false);
    }
    // acc[v] = dot(code ct*16 + v + 8*hi, z[nl]); score = dot - 0.5||c||^2
#pragma unroll
    for (int v = 0; v < 8; ++v) {
      int code = ct * 16 + hi * 8 + v;        // ascending visit order in-lane
      float score = acc[v] - cn_lds[code];
      if (score > best) { best = score; bidx = code; }
    }
  }

  // Combine complementary code halves: lane L <-> lane L^16 (same z column).
  float obest = __shfl_xor(best, 16, 32);
  int   oidx  = __shfl_xor(bidx, 16, 32);
  if (obest > best || (obest == best && oidx < bidx)) { best = obest; bidx = oidx; }
  if (hi == 0) idx_lds[nl] = bidx;
  __syncthreads();

  // Write z_q in (B,C,H,W) layout (coalesced over n, streamed once -> NT
  // store) + accumulate (q - z)^2 for the loss.
  float lsum = 0.f;
  for (int i = t; i < CODE_DIM * ROWS_PB; i += THREADS) {
    int c = i >> 6, n2 = i & 63;
    float q  = codebook[(size_t)idx_lds[n2] * CODE_DIM + c];
    float zv = z_lds[c * 65 + n2];
    __builtin_nontemporal_store(q, &out[zbase + (size_t)c * HW + n2]);
    float d = q - zv;
    lsum += d * d;
  }
  // indices as float, laid out after z_q and loss
  if (t < ROWS_PB)
    __builtin_nontemporal_store((float)idx_lds[t], &out[(size_t)NZ + 1 + n0 + t]);

  // Deterministic block reduction of lsum.
#pragma unroll
  for (int o = 16; o > 0; o >>= 1) lsum += __shfl_down(lsum, o, 32);
  if (lane == 0) wsum[wave] = lsum;
  __syncthreads();
  if (t == 0) partial[blockIdx.x] = (wsum[0] + wsum[1]) + (wsum[2] + wsum[3]);
}

// ---------------------------------------------------------------------------
// Finalize: deterministic reduction of 4096 block partials -> vq_loss.
// loss = (1 + beta) * sum / (256*64*32*32),  beta = 0.25
// ---------------------------------------------------------------------------
__global__ void vq_finalize_44358422233166(const float* __restrict__ partial,
                                           float* __restrict__ out) {
  __shared__ float red[256];
  int t = threadIdx.x;
  float s = 0.f;
  for (int i = t; i < NBLOCKS; i += 256) s += partial[i];   // fixed order
  red[t] = s;
  __syncthreads();
  for (int o = 128; o > 0; o >>= 1) {
    if (t < o) red[t] += red[t + o];
    __syncthreads();
  }
  if (t == 0) out[NZ] = 1.25f * red[0] / (float)NZ;
}

// ---------------------------------------------------------------------------
extern "C" void kernel_launch(void* const* d_in, const int* in_sizes, int n_in,
                              void* d_out, int out_size, void* d_ws, size_t ws_size,
                              hipStream_t stream) {
  const float* z        = (const float*)d_in[0];   // (256, 64, 32, 32) f32
  const float* codebook = (const float*)d_in[1];   // (512, 64) f32
  float* out = (float*)d_out;                      // [z_q | loss | indices]
  float* cnorm   = (float*)d_ws;                   // 512 floats
  float* partial = cnorm + N_CODES;                // 4096 floats

  vq_setup_44358422233166<<<1, N_CODES, 0, stream>>>(codebook, cnorm);
  vq_main_44358422233166<<<NBLOCKS, THREADS, 0, stream>>>(z, codebook, cnorm,
                                                          out, partial);
  vq_finalize_44358422233166<<<1, 256, 0, stream>>>(partial, out);
}